// URRobot_80547816669609
// MI455X (gfx1250) — compile-verified
//
#include <hip/hip_runtime.h>
#include <hip/hip_bf16.h>
#include <math.h>

// ---------------------------------------------------------------------------
// URRobot FK + vertex/normal transform for gfx1250 (MI455X).
//
// Roofline: output is 352 MB f32 (store-bound, ~15us floor @ 23.3 TB/s);
// inputs are 3.7 MB re-read 32x (~117 MB of L2 traffic). Strategy:
//   * K=4 GEMM via V_WMMA_F32_16X16X4_F32:
//       A (16x4) = 4 batches x 4 T_link rows (row 3 = pad)
//       B (4x16) = 16 vertices as columns (coalesced 256B/wave loads)
//       D: lane n = vertex column; d0-d2 / d4-d6 = xyz of two batches
//   * Output stores are NON-TEMPORAL (th:TH_STORE_NT) so the 352 MB stream
//     doesn't evict the L2-resident verts/normals.
//   * btile is the fastest-varying wave index -> 32 consecutive waves read
//     the same vertex chunk (reuse distance ~0).
// ---------------------------------------------------------------------------

typedef __attribute__((ext_vector_type(2))) float v2f;
typedef __attribute__((ext_vector_type(8))) float v8f;

#define NBATCH 128
#define NVERT  16384

// ---------------- Phase 1: forward kinematics (tiny) ----------------

__device__ __forceinline__ void mm4(const float* A, const float* Bm, float* C) {
#pragma unroll
  for (int r = 0; r < 4; ++r)
#pragma unroll
    for (int c = 0; c < 4; ++c)
      C[r*4+c] = A[r*4+0]*Bm[0*4+c] + A[r*4+1]*Bm[1*4+c]
               + A[r*4+2]*Bm[2*4+c] + A[r*4+3]*Bm[3*4+c];
}

// T(x,y,z) @ Rz(yaw) @ Ry(pitch) @ Rx(roll), row-major
__device__ __forceinline__ void t_origin(float rr, float pp, float yw,
                                         float x, float y, float z, float* M) {
  float cy = cosf(yw), sy = sinf(yw);
  float cp = cosf(pp), sp = sinf(pp);
  float cr = cosf(rr), sr = sinf(rr);
  M[0]  = cy*cp;  M[1]  = cy*sp*sr - sy*cr;  M[2]  = cy*sp*cr + sy*sr;  M[3]  = x;
  M[4]  = sy*cp;  M[5]  = sy*sp*sr + cy*cr;  M[6]  = sy*sp*cr - cy*sr;  M[7]  = y;
  M[8]  = -sp;    M[9]  = cp*sr;             M[10] = cp*cr;             M[11] = z;
  M[12] = 0.f;    M[13] = 0.f;               M[14] = 0.f;               M[15] = 1.f;
}

__global__ __launch_bounds__(128) void fk_kernel(const float* __restrict__ pose,
                                                 const float* __restrict__ theta,
                                                 float* __restrict__ Tlink) {
  const int b = threadIdx.x;
  if (b >= NBATCH) return;

  const float KIN[7][6] = {
    {0.f, 0.f, 3.14159265359f, 0.f, 0.f, 0.f},
    {0.f, 0.f, 0.f, 0.f, 0.f, 0.15185f},
    {1.57079632679f, 0.f, 0.f, 0.f, 0.f, 0.f},
    {0.f, 0.f, 0.f, -0.24355f, 0.f, 0.f},
    {0.f, 0.f, 0.f, -0.2132f, 0.f, 0.13105f},
    {1.57079632679f, 0.f, 0.f, 0.f, -0.08535f, 0.f},
    {1.57079632679f, 3.14159265359f, 3.14159265359f, 0.f, 0.0921f, 0.f}};
  const float OFF[7][6] = {
    {0.f, 0.f, 3.14159265359f, 0.f, 0.f, 0.f},
    {0.f, 0.f, 3.14159265359f, 0.f, 0.f, 0.f},
    {1.57079632679f, 0.f, -1.57079632679f, 0.f, 0.f, 0.12f},
    {1.57079632679f, 0.f, -1.57079632679f, 0.f, 0.f, 0.027f},
    {1.57079632679f, 0.f, 0.f, 0.f, 0.f, -0.104f},
    {0.f, 0.f, 0.f, 0.f, 0.f, -0.08535f},
    {1.57079632679f, 0.f, 0.f, 0.f, 0.f, -0.0921f}};

  float T[16], kin[16], off[16], tmp[16];
  t_origin(KIN[0][0], KIN[0][1], KIN[0][2], KIN[0][3], KIN[0][4], KIN[0][5], kin);
  mm4(pose + b * 16, kin, T);

#pragma unroll
  for (int i = 0; i < 7; ++i) {
    if (i > 0) {
      float rz[16], A[16];
      t_origin(KIN[i][0], KIN[i][1], KIN[i][2], KIN[i][3], KIN[i][4], KIN[i][5], kin);
      float th = theta[b * 6 + (i - 1)];
      float ct = cosf(th), st = sinf(th);
      rz[0] = ct;  rz[1] = -st; rz[2] = 0.f; rz[3] = 0.f;
      rz[4] = st;  rz[5] = ct;  rz[6] = 0.f; rz[7] = 0.f;
      rz[8] = 0.f; rz[9] = 0.f; rz[10] = 1.f; rz[11] = 0.f;
      rz[12] = 0.f; rz[13] = 0.f; rz[14] = 0.f; rz[15] = 1.f;
      mm4(kin, rz, A);
      mm4(T, A, tmp);
#pragma unroll
      for (int k = 0; k < 16; ++k) T[k] = tmp[k];
    }
    t_origin(OFF[i][0], OFF[i][1], OFF[i][2], OFF[i][3], OFF[i][4], OFF[i][5], off);
    mm4(T, off, tmp);
    float* o = Tlink + ((size_t)i * NBATCH + b) * 16;
#pragma unroll
    for (int k = 0; k < 16; ++k) o[k] = tmp[k];
  }
}

// ---------------- Phase 2: WMMA vertex/normal transform (store-bound) -------

// 12-byte non-temporal store: b64 + b32, both th:TH_STORE_NT.
__device__ __forceinline__ void store3_nt(float* p, float a, float b, float c) {
  v2f lo = {a, b};
  __builtin_nontemporal_store(lo, (v2f*)p);
  __builtin_nontemporal_store(c, p + 2);
}

__global__ __launch_bounds__(256) void xform_kernel(const float* __restrict__ verts,
                                                    const float* __restrict__ normals,
                                                    const float* __restrict__ Tlink,
                                                    float* __restrict__ out) {
  const int tid  = threadIdx.x;
  const int lane = tid & 31;
  // global wave id: 7 links * 32 vertex-chunks * 32 batch-tiles = 7168 waves.
  // btile varies fastest: 32 consecutive waves share the same vertex chunk.
  const int gw     = blockIdx.x * (blockDim.x >> 5) + (tid >> 5);
  const int btile  = gw & 31;          // 32 tiles of 4 batches   (fastest)
  const int vchunk = (gw >> 5) & 31;   // 32 chunks of 512 vertices
  const int l      = gw >> 10;         // 0..6 (grid sized exactly)

  const int hi = lane >> 4;            // 0: K={0,1}/rows 0-7; 1: K={2,3}/rows 8-15
  const int m  = lane & 15;

  // --- A matrix: 16 rows = 4 batches x 4 rows of T_link (row 3 unused pad)
  const int bi = m >> 2;               // batch within tile
  const int c  = m & 3;                // row of T
  const int b0 = btile * 4;
  const v2f a = *(const v2f*)(Tlink + ((size_t)(l * NBATCH + b0 + bi) * 16 + c * 4 + 2 * hi));

  const float* vl = verts   + (size_t)l * NVERT * 4;
  const float* nl = normals + (size_t)l * NVERT * 4;
  float* ov = out + (size_t)l       * NBATCH * NVERT * 3;  // v_out block
  float* on = out + (size_t)(7 + l) * NBATCH * NVERT * 3;  // n_out block

  const int bA = b0 + 2 * hi;          // this lane's D rows cover batches bA, bA+1
  const v8f cz = {0.f, 0.f, 0.f, 0.f, 0.f, 0.f, 0.f, 0.f};

  for (int t = 0; t < 32; ++t) {
    const int v = vchunk * 512 + t * 16 + m;   // vertex index (column n = m)

    // B matrix: column n = vertex; lane loads components {2hi, 2hi+1} -> float2.
    // 32 lanes together read 256 contiguous bytes (coalesced), default TH (cache).
    const v2f bv = *(const v2f*)(vl + (size_t)v * 4 + 2 * hi);
    const v2f bn = *(const v2f*)(nl + (size_t)v * 4 + 2 * hi);

    v8f d = __builtin_amdgcn_wmma_f32_16x16x4_f32(
        false, a, false, bv, (short)0, cz, false, false);
    v8f e = __builtin_amdgcn_wmma_f32_16x16x4_f32(
        false, a, false, bn, (short)0, cz, false, false);

    // D rows j+8*hi: d0-d2 = batch bA xyz, d4-d6 = batch bA+1 xyz.
    // Non-temporal: don't let the 352 MB output stream evict verts from L2.
    store3_nt(ov + ((size_t)bA       * NVERT + v) * 3, d[0], d[1], d[2]);
    store3_nt(ov + ((size_t)(bA + 1) * NVERT + v) * 3, d[4], d[5], d[6]);
    store3_nt(on + ((size_t)bA       * NVERT + v) * 3, e[0], e[1], e[2]);
    store3_nt(on + ((size_t)(bA + 1) * NVERT + v) * 3, e[4], e[5], e[6]);
  }
}

// ---------------------------------------------------------------------------

extern "C" void kernel_launch(void* const* d_in, const int* in_sizes, int n_in,
                              void* d_out, int out_size, void* d_ws, size_t ws_size,
                              hipStream_t stream) {
  const float* pose    = (const float*)d_in[0];   // [128,4,4]
  const float* theta   = (const float*)d_in[1];   // [128,6]
  const float* verts   = (const float*)d_in[2];   // [7,16384,4]
  const float* normals = (const float*)d_in[3];   // [7,16384,4]
  float* out   = (float*)d_out;                   // [14,128,16384,3]
  float* Tlink = (float*)d_ws;                    // [7,128,4,4] = 57344 B

  fk_kernel<<<1, 128, 0, stream>>>(pose, theta, Tlink);

  // 7 links * 32 vchunks * 32 btiles = 7168 waves; 8 waves/block -> 896 blocks
  xform_kernel<<<896, 256, 0, stream>>>(verts, normals, Tlink, out);
}